// SSM_12034498363525
// MI455X (gfx1250) — compile-verified
//
#include <hip/hip_runtime.h>
#include <math.h>

// ---------------------------------------------------------------------------
// Mamba selective scan for MI455X (gfx1250, wave32).
//   B=2, L=2048, D_INNER=2048, D_STATE=16, DT_RANK=64, N_PROJ=96
// Kernel 1: x_dbl = x @ Wp          (4096x2048)x(2048x96)  -- fp32 WMMA 16x16x4
// Kernel 2: delta = softplus(x_dbl[:, :64] @ Wdt + bias)   -- fp32 WMMA 16x16x4
// Kernels 3-5: chunked parallel scan (first-order linear recurrence composes:
//   128-step chunk -> (A_chunk, B_chunk); 16 chunks combined serially; then
//   each chunk replayed from its entering state). 16x more waves in flight
//   than a flat serial scan -> hides exp/load latency on every SIMD.
// ---------------------------------------------------------------------------

#define BATCH   2
#define SEQLEN  2048
#define D_INNER 2048
#define D_STATE 16
#define DT_RANK 64
#define N_PROJ  (DT_RANK + 2 * D_STATE)   // 96
#define M_TOT   (BATCH * SEQLEN)          // 4096
#define NSTATE_TOT (BATCH * D_INNER * D_STATE)  // 65536 (b,d,n) chains
#define NCHUNK  16
#define CHUNKL  (SEQLEN / NCHUNK)         // 128

typedef float v2f __attribute__((ext_vector_type(2)));
typedef float v8f __attribute__((ext_vector_type(8)));

// --------------------------------------------------------------------------
// GEMM1: Y[M_TOT x 96] = X[M_TOT x 2048] * W[2048 x 96]
// Block = 192 threads (6 waves); wave w owns N-tile 16*w; block owns one
// 16-row M panel. fp32 WMMA 16x16x4.
// --------------------------------------------------------------------------
__global__ void ssm_xproj_wmma(const float* __restrict__ X,
                               const float* __restrict__ W,
                               float* __restrict__ Y) {
  const int lane = threadIdx.x & 31;
  const int wave = threadIdx.x >> 5;      // 0..5
  const int half = lane >> 4;
  const int ml   = lane & 15;
  const int m0   = blockIdx.x << 4;
  const int n0   = wave << 4;

  const float* __restrict__ arow = X + (size_t)(m0 + ml) * 2048;
  v8f acc = {};
  for (int k = 0; k < 2048; k += 4) {
    const int ka = k + 2 * half;
    v2f a, b;
    a.x = arow[ka];
    a.y = arow[ka + 1];
    b.x = W[(size_t)ka * N_PROJ + n0 + ml];
    b.y = W[(size_t)(ka + 1) * N_PROJ + n0 + ml];
    acc = __builtin_amdgcn_wmma_f32_16x16x4_f32(
        false, a, false, b, (short)0, acc, false, false);
  }
#pragma unroll
  for (int r = 0; r < 8; ++r) {
    const int m = m0 + r + 8 * half;
    Y[(size_t)m * N_PROJ + n0 + ml] = acc[r];
  }
}

// --------------------------------------------------------------------------
// GEMM2 + softplus: DT = softplus(XD[:, 0:64] * Wd[64 x 2048] + bias)
// Block = 256 threads (8 waves); wave w owns n0 = 128*bx + 16*w. K=64.
// --------------------------------------------------------------------------
__global__ void ssm_dtproj_wmma(const float* __restrict__ XD,
                                const float* __restrict__ Wd,
                                const float* __restrict__ bias,
                                float* __restrict__ DT) {
  const int lane = threadIdx.x & 31;
  const int wave = threadIdx.x >> 5;      // 0..7
  const int half = lane >> 4;
  const int ml   = lane & 15;
  const int m0   = blockIdx.y << 4;
  const int n0   = (blockIdx.x << 7) + (wave << 4);

  const float* __restrict__ arow = XD + (size_t)(m0 + ml) * N_PROJ;
  v8f acc = {};
  for (int k = 0; k < DT_RANK; k += 4) {
    const int ka = k + 2 * half;
    v2f a, b;
    a.x = arow[ka];
    a.y = arow[ka + 1];
    b.x = Wd[(size_t)ka * D_INNER + n0 + ml];
    b.y = Wd[(size_t)(ka + 1) * D_INNER + n0 + ml];
    acc = __builtin_amdgcn_wmma_f32_16x16x4_f32(
        false, a, false, b, (short)0, acc, false, false);
  }
  const float bn = bias[n0 + ml];
#pragma unroll
  for (int r = 0; r < 8; ++r) {
    const int m = m0 + r + 8 * half;
    float v = acc[r] + bn;
    float sp = (v > 20.0f) ? v : log1pf(__expf(v));   // stable softplus
    DT[(size_t)m * D_INNER + n0 + ml] = sp;
  }
}

// --------------------------------------------------------------------------
// Scan pass 1: per (b,d,n,chunk) run the 128-step recurrence from state 0,
// producing the chunk operator: A_chunk = prod(dA), B_chunk = final state.
// grid = (256, NCHUNK) x 256 threads -> 32768 waves in flight.
// --------------------------------------------------------------------------
__global__ void ssm_scan_pass1(const float* __restrict__ X,
                               const float* __restrict__ XDBL,
                               const float* __restrict__ DT,
                               const float* __restrict__ A_log,
                               float* __restrict__ chunkA,
                               float* __restrict__ chunkB) {
  const int t = blockIdx.x * blockDim.x + threadIdx.x;  // 0..65535
  const int chunk = blockIdx.y;                         // 0..15
  const int n = t & 15;
  const int c = t >> 4;          // channel 0..4095
  const int b = c >> 11;
  const int d = c & 2047;

  const float a2 = -__expf(A_log[d * D_STATE + n]) * 1.44269504088896f;
  float s = 0.0f;
  float p = 1.0f;
  const size_t mbase = (size_t)b * SEQLEN + (size_t)chunk * CHUNKL;
  for (int l = 0; l < CHUNKL; ++l) {
    const size_t m = mbase + l;
    const float dt = DT[m * D_INNER + d];
    const float xv = X[m * D_INNER + d];
    const float Bn = XDBL[m * N_PROJ + DT_RANK + n];
    const float dA = exp2f(dt * a2);          // one v_exp_f32
    s = fmaf(dA, s, dt * Bn * xv);
    p *= dA;
  }
  const size_t i = (size_t)chunk * NSTATE_TOT + t;
  chunkA[i] = p;
  chunkB[i] = s;
}

// --------------------------------------------------------------------------
// Scan pass 2: serially combine the 16 chunk operators per (b,d,n) chain.
// After this, chunkB[chunk] holds the ENTERING state of that chunk.
// 65536 threads, 16 fma steps each -- negligible cost, fully coalesced.
// --------------------------------------------------------------------------
__global__ void ssm_scan_pass2(const float* __restrict__ chunkA,
                               float* __restrict__ chunkB) {
  const int t = blockIdx.x * blockDim.x + threadIdx.x;  // 0..65535
  float s = 0.0f;
#pragma unroll
  for (int ch = 0; ch < NCHUNK; ++ch) {
    const size_t i = (size_t)ch * NSTATE_TOT + t;
    const float Ac = chunkA[i];
    const float Bc = chunkB[i];
    chunkB[i] = s;               // entering state for chunk ch
    s = fmaf(Ac, s, Bc);         // state after chunk ch
  }
}

// --------------------------------------------------------------------------
// Scan pass 3: replay each chunk from its entering state and emit y.
// Lanes [16g..16g+15] hold the 16 states of one channel; <state,C> dot is a
// 4-step shfl_xor butterfly confined to each 16-lane group.
// --------------------------------------------------------------------------
__global__ void ssm_scan_pass3(const float* __restrict__ X,
                               const float* __restrict__ XDBL,
                               const float* __restrict__ DT,
                               const float* __restrict__ A_log,
                               const float* __restrict__ Dv,
                               const float* __restrict__ Sin,   // chunkB
                               float* __restrict__ OUT) {
  const int t = blockIdx.x * blockDim.x + threadIdx.x;  // 0..65535
  const int chunk = blockIdx.y;                         // 0..15
  const int n = t & 15;
  const int c = t >> 4;
  const int b = c >> 11;
  const int d = c & 2047;

  const float a2 = -__expf(A_log[d * D_STATE + n]) * 1.44269504088896f;
  const float Dd = Dv[d];
  float st = Sin[(size_t)chunk * NSTATE_TOT + t];

  const size_t mbase = (size_t)b * SEQLEN + (size_t)chunk * CHUNKL;
  for (int l = 0; l < CHUNKL; ++l) {
    const size_t m = mbase + l;
    const float dt = DT[m * D_INNER + d];
    const float xv = X[m * D_INNER + d];
    const float* __restrict__ bc = XDBL + m * N_PROJ + DT_RANK;  // B | C
    const float Bn = bc[n];
    const float Cn = bc[D_STATE + n];

    const float dA = exp2f(dt * a2);
    st = fmaf(dA, st, dt * Bn * xv);
    float y = st * Cn;
    y += __shfl_xor(y, 1, 32);
    y += __shfl_xor(y, 2, 32);
    y += __shfl_xor(y, 4, 32);
    y += __shfl_xor(y, 8, 32);
    if (n == 0) OUT[m * D_INNER + d] = y + xv * Dd;
  }
}

// --------------------------------------------------------------------------
extern "C" void kernel_launch(void* const* d_in, const int* in_sizes, int n_in,
                              void* d_out, int out_size, void* d_ws, size_t ws_size,
                              hipStream_t stream) {
  const float* x    = (const float*)d_in[0];  // (2, 2048, 2048)
  const float* xpw  = (const float*)d_in[1];  // (2048, 96)
  const float* dtw  = (const float*)d_in[2];  // (64, 2048)
  const float* dtb  = (const float*)d_in[3];  // (2048,)
  const float* alog = (const float*)d_in[4];  // (2048, 16)
  const float* dvec = (const float*)d_in[5];  // (2048,)
  float* out = (float*)d_out;                 // (2, 2048, 2048)

  // workspace layout (fp32):
  //   xdbl   : M_TOT * 96            =  1.5 MB
  //   delta  : M_TOT * 2048          = 32.0 MB
  //   chunkA : NCHUNK * NSTATE_TOT   =  4.0 MB
  //   chunkB : NCHUNK * NSTATE_TOT   =  4.0 MB
  char* ws = (char*)d_ws;
  float* xdbl   = (float*)ws;
  ws += (size_t)M_TOT * N_PROJ * sizeof(float);
  float* delta  = (float*)ws;
  ws += (size_t)M_TOT * D_INNER * sizeof(float);
  float* chunkA = (float*)ws;
  ws += (size_t)NCHUNK * NSTATE_TOT * sizeof(float);
  float* chunkB = (float*)ws;

  // GEMM1: 256 blocks x 192 threads (6 waves, one per 16-col N tile of 96)
  ssm_xproj_wmma<<<dim3(M_TOT / 16), dim3(192), 0, stream>>>(x, xpw, xdbl);

  // GEMM2: grid (2048/128, 4096/16) x 256 threads (8 waves, 16 cols each)
  ssm_dtproj_wmma<<<dim3(D_INNER / 128, M_TOT / 16), dim3(256), 0, stream>>>(
      xdbl, dtw, dtb, delta);

  // Chunked scan: pass1 (chunk operators) -> pass2 (combine) -> pass3 (replay)
  ssm_scan_pass1<<<dim3(NSTATE_TOT / 256, NCHUNK), dim3(256), 0, stream>>>(
      x, xdbl, delta, alog, chunkA, chunkB);
  ssm_scan_pass2<<<dim3(NSTATE_TOT / 256), dim3(256), 0, stream>>>(
      chunkA, chunkB);
  ssm_scan_pass3<<<dim3(NSTATE_TOT / 256, NCHUNK), dim3(256), 0, stream>>>(
      x, xdbl, delta, alog, dvec, chunkB, out);
}